// TransientGenerator_7387343749605
// MI455X (gfx1250) — compile-verified
//
#include <hip/hip_runtime.h>
#include <stdint.h>
#include <math.h>

// Problem constants (fixed by the reference)
#define SR    16000
#define BATCH 64
#define NT    256      // MAX_TRANSIENTS per batch (sorted by start time)
#define TLEN  1600     // TRANSIENT_SAMPLES
#define NTPL  20       // N_TRANSIENTS (template rows)

#define TPB   256      // threads per block
#define SPT   4        // output samples per thread (float4)
#define SPB   (TPB * SPT)  // samples per block = 1024

// ---- CDNA5 async global->LDS staging -------------------------------------

// LDS byte offset of a __shared__ object (addrspacecast flat -> local).
__device__ __forceinline__ uint32_t lds_off(const void* p) {
  return (uint32_t)(uintptr_t)(const __attribute__((address_space(3))) void*)p;
}

// global_load_async_to_lds_b32: per-lane 4B copy, GV addressing mode.
// VDST = VGPR with LDS byte offset, VADDR = 64-bit global address, saddr=off.
__device__ __forceinline__ void async_ld_b32(uint32_t lds_dst, const void* gsrc) {
  asm volatile("global_load_async_to_lds_b32 %0, %1, off"
               :: "v"(lds_dst), "v"(gsrc)
               : "memory");
}

__device__ __forceinline__ void wait_async0() {
#if __has_builtin(__builtin_amdgcn_s_wait_asynccnt)
  __builtin_amdgcn_s_wait_asynccnt(0);
#else
  asm volatile("s_wait_asynccnt 0" ::: "memory");
#endif
}

// ---- kernel ---------------------------------------------------------------

__global__ __launch_bounds__(TPB)
void transient_gather_kernel(const float* __restrict__ timings,
                             const int*   __restrict__ ids,
                             const float* __restrict__ gains,
                             const float* __restrict__ tmpl,
                             float*       __restrict__ out,
                             int N /* audio_length */) {
  __shared__ int   s_start[NT];  // start sample (sorted, non-decreasing)
  __shared__ float s_gain[NT];   // gain, zeroed if transient invalid
  __shared__ int   s_row[NT];    // template row offset = clipped_id * TLEN

  const int i = threadIdx.x;     // 0..255 == transient slot for staging
  const int b = blockIdx.y;      // batch
  const int base = b * NT + i;

  // Stage this batch's transient metadata into LDS via async copy engine.
  async_ld_b32(lds_off(&s_start[i]), timings + base);  // raw f32 bits
  async_ld_b32(lds_off(&s_gain[i]),  gains   + base);
  async_ld_b32(lds_off(&s_row[i]),   ids     + base);
  wait_async0();
  __syncthreads();

  // In-place conversion (thread i owns element i; no cross-thread hazard).
  {
    const float tsec = __int_as_float(s_start[i]);
    const int   st   = (int)floorf(tsec * (float)SR);  // == jnp.floor(t*SR).astype(i32)
    const int   id   = s_row[i];
    const float g    = s_gain[i];
    const bool valid = (g > 0.0f) && (id < NTPL) && (st < N);
    int idc = id < 0 ? 0 : (id > (NTPL - 1) ? (NTPL - 1) : id);
    s_start[i] = st;
    s_row[i]   = idc * TLEN;
    s_gain[i]  = valid ? g : 0.0f;
  }
  __syncthreads();

  const int pos0 = blockIdx.x * SPB + i * SPT;
  if (pos0 >= N) return;

  // Transient k contributes to [pos0, pos0+3] iff
  //   start[k] >= pos0 - (TLEN-1)   and   start[k] <= pos0 + 3.
  // s_start is sorted -> two binary searches over LDS.
  const int keyLo = pos0 - (TLEN - 1);
  int lo = 0, hi = NT;
  while (lo < hi) {                       // lower_bound(keyLo)
    int m = (lo + hi) >> 1;
    if (s_start[m] < keyLo) lo = m + 1; else hi = m;
  }
  const int first = lo;
  const int keyHi = pos0 + (SPT - 1);
  hi = NT;                                // upper_bound(keyHi)
  while (lo < hi) {
    int m = (lo + hi) >> 1;
    if (s_start[m] <= keyHi) lo = m + 1; else hi = m;
  }
  const int last = lo;

  float4 acc = make_float4(0.f, 0.f, 0.f, 0.f);
  for (int k = first; k < last; ++k) {
    const float g    = s_gain[k];
    const int   off0 = pos0 - s_start[k];        // in [-3, TLEN-1]
    const float* t   = tmpl + s_row[k];          // 128 KB table, L2-resident
    const int o0 = off0, o1 = off0 + 1, o2 = off0 + 2, o3 = off0 + 3;
    if ((unsigned)o0 < (unsigned)TLEN) acc.x = fmaf(g, t[o0], acc.x);
    if ((unsigned)o1 < (unsigned)TLEN) acc.y = fmaf(g, t[o1], acc.y);
    if ((unsigned)o2 < (unsigned)TLEN) acc.z = fmaf(g, t[o2], acc.z);
    if ((unsigned)o3 < (unsigned)TLEN) acc.w = fmaf(g, t[o3], acc.w);
  }

  // Exactly-once store; no zero-init pass needed.
  if (pos0 + SPT <= N) {
    *reinterpret_cast<float4*>(out + (size_t)b * N + pos0) = acc;
  } else {
    const float a[4] = {acc.x, acc.y, acc.z, acc.w};
    for (int j = 0; j < SPT && pos0 + j < N; ++j)
      out[(size_t)b * N + pos0 + j] = a[j];
  }
}

// ---- launcher -------------------------------------------------------------

extern "C" void kernel_launch(void* const* d_in, const int* in_sizes, int n_in,
                              void* d_out, int out_size, void* d_ws, size_t ws_size,
                              hipStream_t stream) {
  const float* timings = (const float*)d_in[0];  // [64,256] f32, sorted per row
  const int*   ids     = (const int*)  d_in[1];  // [64,256] i32
  const float* gains   = (const float*)d_in[2];  // [64,256] f32
  const float* tmpl    = (const float*)d_in[3];  // [20,1600] f32
  float* out = (float*)d_out;                    // [64, N] f32

  const int N = out_size / BATCH;                // audio_length (160000)
  dim3 grid((N + SPB - 1) / SPB, BATCH);
  transient_gather_kernel<<<grid, TPB, 0, stream>>>(timings, ids, gains, tmpl, out, N);
}